// V2XGraphPredictor_23347442221671
// MI455X (gfx1250) — compile-verified
//
#include <hip/hip_runtime.h>
#include <hip/hip_bf16.h>
#include <stdint.h>

// ---------------------------------------------------------------------------
// Multi-head GAT layer on gfx1250 (MI455X), fp32 end-to-end using
// V_WMMA_F32_16X16X4_F32 (exact fp32 multiply, fp32 accumulate), with
// GLOBAL_LOAD_ASYNC_TO_LDS_B128 staging (ASYNCcnt) for all LDS tiles.
// B=8, N=1024, D=256, H=8, hd=32, Dff=512.
// ---------------------------------------------------------------------------

static constexpr int BATCH = 8;
static constexpr int NNODE = 1024;
static constexpr int DMODEL = 256;
static constexpr int NHEAD = 8;
static constexpr int HDIM = 32;
static constexpr int DFF = 512;

typedef __attribute__((ext_vector_type(2))) float v2f;
typedef __attribute__((ext_vector_type(8))) float v8f;

__device__ __forceinline__ v8f wmma_f32x4(v2f a, v2f b, v8f c) {
  // D = A(16x4) * B(4x16) + C(16x16), all fp32.
  // 8 args: (neg_a, A, neg_b, B, c_mod, C, reuse_a, reuse_b)
  return __builtin_amdgcn_wmma_f32_16x16x4_f32(false, a, false, b, (short)0, c,
                                               false, false);
}

__device__ __forceinline__ float wave_sum(float v) {
#pragma unroll
  for (int o = 16; o > 0; o >>= 1) v += __shfl_xor(v, o, 32);
  return v;
}
__device__ __forceinline__ float wave_max(float v) {
#pragma unroll
  for (int o = 16; o > 0; o >>= 1) v = fmaxf(v, __shfl_xor(v, o, 32));
  return v;
}

// --- CDNA5 async global->LDS copy (no VGPR round trip, tracked by ASYNCcnt) --
__device__ __forceinline__ void async_b128_to_lds(uint32_t lds_byte_off,
                                                  const void* gptr) {
  asm volatile("global_load_async_to_lds_b128 %0, %1, off"
               :
               : "v"(lds_byte_off), "v"(gptr)
               : "memory");
}
__device__ __forceinline__ void wait_async_then_barrier() {
  asm volatile("s_wait_asynccnt 0" ::: "memory");
  __syncthreads();
}
__device__ __forceinline__ uint32_t lds_offset_of(const void* shared_ptr) {
  // Low 32 bits of a generic LDS-aperture address are the LDS byte offset.
  return (uint32_t)(uintptr_t)shared_ptr;
}

// Stage ROWS x COLS fp32 tile (global row stride = src_stride floats) into a
// padded LDS tile via async B128 copies.  COLS, PAD*4, stride must keep 16B
// alignment (true for all uses below).
template <int ROWS, int COLS, int PAD>
__device__ __forceinline__ void stage_tile_async(float (&dst)[ROWS][COLS + PAD],
                                                 const float* __restrict__ src,
                                                 int src_stride) {
  const uint32_t base = lds_offset_of(&dst[0][0]);
  constexpr int CHUNKS = ROWS * (COLS / 4);  // 16-byte chunks
  constexpr int CPR = COLS / 4;              // chunks per row
  for (int c = threadIdx.x; c < CHUNKS; c += 256) {
    const int r = c / CPR;
    const int q = c % CPR;
    async_b128_to_lds(base + (uint32_t)(r * (COLS + PAD) + q * 4) * 4u,
                      src + (size_t)r * src_stride + q * 4);
  }
}

// ---------------------------------------------------------------------------
// Kernel 1: Wh[b, i, h*32+j] = sum_d h[b,i,d] * W[h,d,j]
// Block = 256 thr (8 waves) computes a 16-row x 256-col tile.
// ---------------------------------------------------------------------------
__global__ __launch_bounds__(256) void k_project(const float* __restrict__ h,
                                                 const float* __restrict__ W,
                                                 float* __restrict__ Wh) {
  __shared__ __align__(16) float sA[16][DMODEL + 4];
  const int b = blockIdx.y;
  const int row0 = blockIdx.x * 16;
  stage_tile_async<16, DMODEL, 4>(sA, h + (size_t)(b * NNODE + row0) * DMODEL,
                                  DMODEL);
  __builtin_prefetch(W, 0, 1);
  wait_async_then_barrier();

  const int wave = threadIdx.x >> 5, lane = threadIdx.x & 31;
  const int lrow = lane & 15, khalf = lane >> 4;
#pragma unroll
  for (int sub = 0; sub < 2; ++sub) {
    const int col = (wave * 2 + sub) * 16 + lrow;  // 0..255
    const int head = col >> 5;
    const int jloc = col & 31;
    v8f acc = {0.f, 0.f, 0.f, 0.f, 0.f, 0.f, 0.f, 0.f};
    for (int k0 = 0; k0 < DMODEL; k0 += 4) {
      const int k = k0 + 2 * khalf;
      v2f av, bv;
      av.x = sA[lrow][k];
      av.y = sA[lrow][k + 1];
      bv.x = W[(size_t)(head * DMODEL + k) * HDIM + jloc];
      bv.y = W[(size_t)(head * DMODEL + k + 1) * HDIM + jloc];
      acc = wmma_f32x4(av, bv, acc);
    }
#pragma unroll
    for (int r = 0; r < 8; ++r) {
      const int row = row0 + r + 8 * khalf;
      Wh[(size_t)(b * NNODE + row) * DMODEL + col] = acc[r];
    }
  }
}

// ---------------------------------------------------------------------------
// Kernel 2: e_src/e_dst, one thread per (b,h,i).
// ---------------------------------------------------------------------------
__global__ __launch_bounds__(256) void k_escore(const float* __restrict__ Wh,
                                                const float* __restrict__ avec,
                                                float* __restrict__ e_src,
                                                float* __restrict__ e_dst) {
  const int idx = blockIdx.x * 256 + threadIdx.x;  // = ((b*H)+h)*N + i
  const int i = idx & (NNODE - 1);
  const int hh = (idx >> 10) & (NHEAD - 1);
  const int b = idx >> 13;
  const float* wp = Wh + (size_t)(b * NNODE + i) * DMODEL + hh * HDIM;
  const float* ap = avec + hh * 2 * HDIM;
  float s1 = 0.f, s2 = 0.f;
#pragma unroll
  for (int j = 0; j < HDIM; ++j) {
    const float w = wp[j];
    s1 += w * ap[j];
    s2 += w * ap[HDIM + j];
  }
  e_src[idx] = s1;
  e_dst[idx] = s2;
}

// ---------------------------------------------------------------------------
// Kernel 3: masked softmax stats. Block per (b,i); adj row async-staged in
// LDS; wave w handles head w; two passes (max, then sum-exp).
// ---------------------------------------------------------------------------
__global__ __launch_bounds__(256) void k_stats(const int* __restrict__ adj,
                                               const float* __restrict__ e_src,
                                               const float* __restrict__ e_dst,
                                               float* __restrict__ mrow,
                                               float* __restrict__ invs) {
  __shared__ __align__(16) int sAdj[NNODE];
  const int b = blockIdx.y, i = blockIdx.x;
  {
    const uint32_t base = lds_offset_of(&sAdj[0]);
    const int* src = adj + (size_t)(b * NNODE + i) * NNODE;
    const int c = threadIdx.x;  // 256 chunks of 16B cover 1024 ints
    async_b128_to_lds(base + (uint32_t)c * 16u, src + c * 4);
  }
  wait_async_then_barrier();

  const int hh = threadIdx.x >> 5, lane = threadIdx.x & 31;
  const int base = (b * NHEAD + hh) * NNODE;
  const float es = e_src[base + i];
  const float* edp = e_dst + base;

  float m = -3.0e38f;
  for (int j = lane; j < NNODE; j += 32) {
    if (sAdj[j]) {
      float e = es + edp[j];
      e = e > 0.f ? e : 0.2f * e;
      m = fmaxf(m, e);
    }
  }
  m = wave_max(m);

  float s = 0.f;
  for (int j = lane; j < NNODE; j += 32) {
    if (sAdj[j]) {
      float e = es + edp[j];
      e = e > 0.f ? e : 0.2f * e;
      s += __expf(e - m);
    }
  }
  s = wave_sum(s);
  if (lane == 0) {
    mrow[base + i] = (s > 0.f) ? m : 0.f;        // avoid inf-inf later
    invs[base + i] = (s > 0.f) ? 1.f / s : 0.f;  // fully-masked row -> alpha 0
  }
}

// ---------------------------------------------------------------------------
// Kernel 4: attention aggregation + residual + LayerNorm1 -> hcat (d_out).
// Block per (b, 16-row tile); wave w = head w. Per K-step of 4 neighbors,
// each lane materializes 2 alpha values in the exact 16x4 fp32 A layout,
// then two WMMAs cover the 32-wide head output.
// ---------------------------------------------------------------------------
__global__ __launch_bounds__(256) void k_attn(
    const int* __restrict__ adj, const float* __restrict__ e_src,
    const float* __restrict__ e_dst, const float* __restrict__ mrow,
    const float* __restrict__ invs, const float* __restrict__ Wh,
    const float* __restrict__ h, const float* __restrict__ g1,
    const float* __restrict__ bt1, float* __restrict__ hcat) {
  __shared__ __align__(16) float sOut[16][DMODEL + 4];
  const int b = blockIdx.y, row0 = blockIdx.x * 16;
  const int wave = threadIdx.x >> 5, lane = threadIdx.x & 31;
  const int lrow = lane & 15, khalf = lane >> 4;
  const int hh = wave;

  const int base = (b * NHEAD + hh) * NNODE;
  const int rowi = row0 + lrow;
  const float es = e_src[base + rowi];
  const float mm = mrow[base + rowi];
  const float iv = invs[base + rowi];
  const int* adjr = adj + (size_t)(b * NNODE + rowi) * NNODE;
  const float* edp = e_dst + base;
  const float* whp = Wh + (size_t)b * NNODE * DMODEL + hh * HDIM;

  v8f acc0 = {0.f, 0.f, 0.f, 0.f, 0.f, 0.f, 0.f, 0.f};
  v8f acc1 = {0.f, 0.f, 0.f, 0.f, 0.f, 0.f, 0.f, 0.f};

  for (int k0 = 0; k0 < NNODE; k0 += 4) {
    const int j = k0 + 2 * khalf;  // even -> 8B aligned pair loads
    const int2 msk = *(const int2*)(adjr + j);
    const float2 ed = *(const float2*)(edp + j);
    float a0 = 0.f, a1 = 0.f;
    if (msk.x) {
      float e = es + ed.x;
      e = e > 0.f ? e : 0.2f * e;
      a0 = __expf(e - mm) * iv;
    }
    if (msk.y) {
      float e = es + ed.y;
      e = e > 0.f ? e : 0.2f * e;
      a1 = __expf(e - mm) * iv;
    }
    v2f av;
    av.x = a0;
    av.y = a1;
    const float* w0 = whp + (size_t)j * DMODEL;
    v2f b0, b1;
    b0.x = w0[lrow];
    b0.y = w0[DMODEL + lrow];
    b1.x = w0[16 + lrow];
    b1.y = w0[DMODEL + 16 + lrow];
    acc0 = wmma_f32x4(av, b0, acc0);
    acc1 = wmma_f32x4(av, b1, acc1);
  }

#pragma unroll
  for (int r = 0; r < 8; ++r) {
    const int ml = r + 8 * khalf;
    sOut[ml][hh * HDIM + lrow] = acc0[r];
    sOut[ml][hh * HDIM + 16 + lrow] = acc1[r];
  }
  __syncthreads();

  // residual + LayerNorm1; wave handles 2 rows, lane covers 8 columns.
#pragma unroll
  for (int t = 0; t < 2; ++t) {
    const int lr = wave * 2 + t;
    const size_t grow = (size_t)(b * NNODE + row0 + lr) * DMODEL;
    float x[8];
    float sum = 0.f;
#pragma unroll
    for (int q = 0; q < 8; ++q) {
      const int c = lane + 32 * q;
      x[q] = sOut[lr][c] + h[grow + c];
      sum += x[q];
    }
    sum = wave_sum(sum);
    const float mu = sum * (1.f / DMODEL);
    float var = 0.f;
#pragma unroll
    for (int q = 0; q < 8; ++q) {
      const float d = x[q] - mu;
      var += d * d;
    }
    var = wave_sum(var) * (1.f / DMODEL);
    const float rs = rsqrtf(var + 1e-5f);
#pragma unroll
    for (int q = 0; q < 8; ++q) {
      const int c = lane + 32 * q;
      hcat[grow + c] = (x[q] - mu) * rs * g1[c] + bt1[c];
    }
  }
}

// ---------------------------------------------------------------------------
// Kernel 5: inter = relu(hcat @ W1 + b1)   (256 -> 512)
// ---------------------------------------------------------------------------
__global__ __launch_bounds__(256) void k_ffn1(const float* __restrict__ hcat,
                                              const float* __restrict__ W1,
                                              const float* __restrict__ bias1,
                                              float* __restrict__ inter) {
  __shared__ __align__(16) float sA[16][DMODEL + 4];
  const int b = blockIdx.y, row0 = blockIdx.x * 16;
  stage_tile_async<16, DMODEL, 4>(
      sA, hcat + (size_t)(b * NNODE + row0) * DMODEL, DMODEL);
  __builtin_prefetch(W1, 0, 1);
  wait_async_then_barrier();

  const int wave = threadIdx.x >> 5, lane = threadIdx.x & 31;
  const int lrow = lane & 15, khalf = lane >> 4;
#pragma unroll
  for (int s = 0; s < 4; ++s) {
    const int col = (wave * 4 + s) * 16 + lrow;  // 0..511
    v8f acc = {0.f, 0.f, 0.f, 0.f, 0.f, 0.f, 0.f, 0.f};
    for (int k0 = 0; k0 < DMODEL; k0 += 4) {
      const int k = k0 + 2 * khalf;
      v2f av, bv;
      av.x = sA[lrow][k];
      av.y = sA[lrow][k + 1];
      bv.x = W1[(size_t)k * DFF + col];
      bv.y = W1[(size_t)(k + 1) * DFF + col];
      acc = wmma_f32x4(av, bv, acc);
    }
    const float bb = bias1[col];
#pragma unroll
    for (int r = 0; r < 8; ++r) {
      const float v = acc[r] + bb;
      inter[(size_t)(b * NNODE + row0 + r + 8 * khalf) * DFF + col] =
          v > 0.f ? v : 0.f;
    }
  }
}

// ---------------------------------------------------------------------------
// Kernel 6: out = LN2(hcat + inter @ W2 + b2)  (512 -> 256), in-place on d_out.
// ---------------------------------------------------------------------------
__global__ __launch_bounds__(256) void k_ffn2(const float* __restrict__ inter,
                                              const float* __restrict__ W2,
                                              const float* __restrict__ bias2,
                                              const float* __restrict__ g2,
                                              const float* __restrict__ bt2,
                                              float* __restrict__ io) {
  __shared__ __align__(16) float sA[16][DFF + 4];
  __shared__ __align__(16) float sOut[16][DMODEL + 4];
  const int b = blockIdx.y, row0 = blockIdx.x * 16;
  stage_tile_async<16, DFF, 4>(sA, inter + (size_t)(b * NNODE + row0) * DFF,
                               DFF);
  __builtin_prefetch(W2, 0, 1);
  wait_async_then_barrier();

  const int wave = threadIdx.x >> 5, lane = threadIdx.x & 31;
  const int lrow = lane & 15, khalf = lane >> 4;
#pragma unroll
  for (int s = 0; s < 2; ++s) {
    const int col = (wave * 2 + s) * 16 + lrow;  // 0..255
    v8f acc = {0.f, 0.f, 0.f, 0.f, 0.f, 0.f, 0.f, 0.f};
    for (int k0 = 0; k0 < DFF; k0 += 4) {
      const int k = k0 + 2 * khalf;
      v2f av, bv;
      av.x = sA[lrow][k];
      av.y = sA[lrow][k + 1];
      bv.x = W2[(size_t)k * DMODEL + col];
      bv.y = W2[(size_t)(k + 1) * DMODEL + col];
      acc = wmma_f32x4(av, bv, acc);
    }
    const float bb = bias2[col];
#pragma unroll
    for (int r = 0; r < 8; ++r) sOut[r + 8 * khalf][col] = acc[r] + bb;
  }
  __syncthreads();

#pragma unroll
  for (int t = 0; t < 2; ++t) {
    const int lr = wave * 2 + t;
    const size_t grow = (size_t)(b * NNODE + row0 + lr) * DMODEL;
    float x[8];
    float sum = 0.f;
#pragma unroll
    for (int q = 0; q < 8; ++q) {
      const int c = lane + 32 * q;
      x[q] = sOut[lr][c] + io[grow + c];  // residual with hcat
      sum += x[q];
    }
    sum = wave_sum(sum);
    const float mu = sum * (1.f / DMODEL);
    float var = 0.f;
#pragma unroll
    for (int q = 0; q < 8; ++q) {
      const float d = x[q] - mu;
      var += d * d;
    }
    var = wave_sum(var) * (1.f / DMODEL);
    const float rs = rsqrtf(var + 1e-5f);
#pragma unroll
    for (int q = 0; q < 8; ++q) {
      const int c = lane + 32 * q;
      io[grow + c] = (x[q] - mu) * rs * g2[c] + bt2[c];
    }
  }
}

// ---------------------------------------------------------------------------
extern "C" void kernel_launch(void* const* d_in, const int* in_sizes, int n_in,
                              void* d_out, int out_size, void* d_ws,
                              size_t ws_size, hipStream_t stream) {
  const float* h = (const float*)d_in[0];
  const int* adj = (const int*)d_in[1];
  const float* W = (const float*)d_in[2];
  const float* avec = (const float*)d_in[3];
  const float* ln1g = (const float*)d_in[4];
  const float* ln1b = (const float*)d_in[5];
  const float* W1 = (const float*)d_in[6];
  const float* b1 = (const float*)d_in[7];
  const float* W2 = (const float*)d_in[8];
  const float* b2 = (const float*)d_in[9];
  const float* ln2g = (const float*)d_in[10];
  const float* ln2b = (const float*)d_in[11];
  float* out = (float*)d_out;

  // Workspace layout (floats): Wh (2M) | e_src | e_dst | m | 1/s (64K each) |
  // inter (4M)  => ~25 MB total.
  float* Wh = (float*)d_ws;
  float* e_src = Wh + (size_t)BATCH * NNODE * DMODEL;
  float* e_dst = e_src + (size_t)BATCH * NHEAD * NNODE;
  float* mrow = e_dst + (size_t)BATCH * NHEAD * NNODE;
  float* invs = mrow + (size_t)BATCH * NHEAD * NNODE;
  float* inter = invs + (size_t)BATCH * NHEAD * NNODE;

  const dim3 blk(256);
  k_project<<<dim3(NNODE / 16, BATCH), blk, 0, stream>>>(h, W, Wh);
  k_escore<<<dim3((BATCH * NHEAD * NNODE) / 256), blk, 0, stream>>>(Wh, avec,
                                                                    e_src,
                                                                    e_dst);
  k_stats<<<dim3(NNODE, BATCH), blk, 0, stream>>>(adj, e_src, e_dst, mrow,
                                                  invs);
  k_attn<<<dim3(NNODE / 16, BATCH), blk, 0, stream>>>(adj, e_src, e_dst, mrow,
                                                      invs, Wh, h, ln1g, ln1b,
                                                      out);
  k_ffn1<<<dim3(NNODE / 16, BATCH), blk, 0, stream>>>(out, W1, b1, inter);
  k_ffn2<<<dim3(NNODE / 16, BATCH), blk, 0, stream>>>(inter, W2, b2, ln2g,
                                                      ln2b, out);
}